// SpatialTransformer_52518860096006
// MI455X (gfx1250) — compile-verified
//
#include <hip/hip_runtime.h>

// 3-D spatial transformer (reflect-pad trilinear warp), MI455X / gfx1250.
// Memory-bound: ~224 MiB HBM traffic @ 23.3 TB/s => ~10 us floor.
// gfx1250 specifics used:
//  - global_load_async_to_lds_b32 (ASYNCcnt) double-buffers the streamed flow
//    into LDS, overlapped with gather+blend compute; drained with
//    s_wait_asynccnt, buffer reuse fenced with s_wait_dscnt.
//  - th:TH_LOAD_NT on the flow stream and non-temporal output stores keep the
//    192 MB L2 dedicated to the 64 MiB image, which absorbs the 8x gather
//    amplification.

#define TPB 256
#define TILES_PER_BLOCK 8
#define NSP (128u * 128u * 128u) /* 2^21 spatial points per batch */
#define NB 4u
#define NC 2u

__device__ __forceinline__ float reflect128(float v) {
  // Matches the reference exactly: twice = 254;
  //   a = |v| % 254  (JAX floor-mod: a - floor(a/254)*254)
  //   return a > 127 ? 254 - a : a
  // Branchless (no fmodf slow-path loop / EXEC divergence).
  float a = fabsf(v);
  a = a - floorf(a / 254.0f) * 254.0f;
  return a > 127.0f ? 254.0f - a : a;
}

// Issue one tile (3 components x 256 floats) of flow into LDS via the CDNA5
// async global->LDS path. One instruction per component => ASYNCcnt += 3.
// Flow is streamed exactly once -> non-temporal hint keeps it out of L2.
#define ISSUE_TILE(T_IDX, BUF)                                                 \
  do {                                                                         \
    unsigned s_ = blockBase + (unsigned)(T_IDX) * TPB + tid;                   \
    unsigned r_ = s_ & (NSP - 1u);                                             \
    unsigned b_ = s_ >> 21;                                                    \
    unsigned e0_ = b_ * (3u * NSP) + r_; /* elem offset of flow comp 0 */      \
    asm volatile(                                                              \
        "global_load_async_to_lds_b32 %0, %1, %2 offset:0 th:TH_LOAD_NT" ::   \
            "v"(ldsAddr[BUF][0]),                                              \
        "v"(e0_ * 4u), "s"(flow)                                               \
        : "memory");                                                           \
    asm volatile(                                                              \
        "global_load_async_to_lds_b32 %0, %1, %2 offset:0 th:TH_LOAD_NT" ::   \
            "v"(ldsAddr[BUF][1]),                                              \
        "v"((e0_ + NSP) * 4u), "s"(flow)                                       \
        : "memory");                                                           \
    asm volatile(                                                              \
        "global_load_async_to_lds_b32 %0, %1, %2 offset:0 th:TH_LOAD_NT" ::   \
            "v"(ldsAddr[BUF][2]),                                              \
        "v"((e0_ + 2u * NSP) * 4u), "s"(flow)                                  \
        : "memory");                                                           \
  } while (0)

__global__ __launch_bounds__(TPB) void st3d_trilinear_kernel(
    const float* __restrict__ img, const float* __restrict__ flow,
    float* __restrict__ out) {
  __shared__ float sflow[2][3][TPB];
  const unsigned tid = threadIdx.x;
  const unsigned blockBase = blockIdx.x * (TPB * TILES_PER_BLOCK);

  // Per-lane LDS byte addresses for the async destinations (low 32 bits of
  // the flat address == LDS offset per the gfx1250 aperture mapping).
  unsigned ldsAddr[2][3];
#pragma unroll
  for (int bf = 0; bf < 2; ++bf)
#pragma unroll
    for (int k = 0; k < 3; ++k)
      ldsAddr[bf][k] = (unsigned)(size_t)(&sflow[bf][k][tid]);

  ISSUE_TILE(0, 0); // prefetch tile 0

#pragma unroll 2
  for (int i = 0; i < TILES_PER_BLOCK; ++i) {
    const int bf = i & 1;

    // Drain the async copy for the current tile (only one tile in flight).
    asm volatile("s_wait_asynccnt 0" ::: "memory");

    // Plain LDS reads -> ds_load_b32.
    float fz = sflow[bf][0][tid];
    float fy = sflow[bf][1][tid];
    float fx = sflow[bf][2][tid];

    // Make sure the LDS reads above retired before the buffer can be
    // re-targeted by a later async write (buffer bf is reused at tile i+2,
    // whose issue happens after this fence in iteration i+1).
    asm volatile("s_wait_dscnt 0" ::: "memory");

    if (i + 1 < TILES_PER_BLOCK) {
      ISSUE_TILE(i + 1, (i + 1) & 1); // overlap next tile's copy with compute
    }

    const unsigned s = blockBase + (unsigned)i * TPB + tid;
    const unsigned r = s & (NSP - 1u);
    const unsigned b = s >> 21;

    const int z = (int)(r >> 14);
    const int y = (int)((r >> 7) & 127u);
    const int x = (int)(r & 127u);

    const float cz = reflect128((float)z + fz);
    const float cy = reflect128((float)y + fy);
    const float cx = reflect128((float)x + fx);

    const float z0f = floorf(cz), y0f = floorf(cy), x0f = floorf(cx);
    const float wz = cz - z0f, wy = cy - y0f, wx = cx - x0f;

    int z0 = (int)z0f; z0 = z0 < 0 ? 0 : (z0 > 127 ? 127 : z0);
    int y0 = (int)y0f; y0 = y0 < 0 ? 0 : (y0 > 127 ? 127 : y0);
    int x0 = (int)x0f; x0 = x0 < 0 ? 0 : (x0 > 127 ? 127 : x0);
    const int z1 = z0 < 127 ? z0 + 1 : 127;
    const int y1 = y0 < 127 ? y0 + 1 : 127;
    const int x1 = x0 < 127 ? x0 + 1 : 127;

    const float uz = 1.0f - wz, uy = 1.0f - wy, ux = 1.0f - wx;
    const float wzy00 = uz * uy, wzy01 = uz * wy;
    const float wzy10 = wz * uy, wzy11 = wz * wy;
    const float w000 = wzy00 * ux, w001 = wzy00 * wx;
    const float w010 = wzy01 * ux, w011 = wzy01 * wx;
    const float w100 = wzy10 * ux, w101 = wzy10 * wx;
    const float w110 = wzy11 * ux, w111 = wzy11 * wx;

    const unsigned r00 = ((unsigned)(z0 * 128 + y0)) << 7;
    const unsigned r01 = ((unsigned)(z0 * 128 + y1)) << 7;
    const unsigned r10 = ((unsigned)(z1 * 128 + y0)) << 7;
    const unsigned r11 = ((unsigned)(z1 * 128 + y1)) << 7;

    const float* ib = img + (size_t)b * (NC * NSP);
    float res[NC];
#pragma unroll
    for (int c = 0; c < (int)NC; ++c) {
      const float* p = ib + (size_t)c * NSP; // image tile is L2-resident
      float acc = w000 * p[r00 + (unsigned)x0];
      acc = fmaf(w001, p[r00 + (unsigned)x1], acc);
      acc = fmaf(w010, p[r01 + (unsigned)x0], acc);
      acc = fmaf(w011, p[r01 + (unsigned)x1], acc);
      acc = fmaf(w100, p[r10 + (unsigned)x0], acc);
      acc = fmaf(w101, p[r10 + (unsigned)x1], acc);
      acc = fmaf(w110, p[r11 + (unsigned)x0], acc);
      acc = fmaf(w111, p[r11 + (unsigned)x1], acc);
      res[c] = acc;
    }

    // Streamed output: non-temporal so it doesn't evict the cached image.
    const size_t ob = (size_t)b * (NC * NSP) + r;
    __builtin_nontemporal_store(res[0], &out[ob]);
    __builtin_nontemporal_store(res[1], &out[ob + NSP]);
  }
}

extern "C" void kernel_launch(void* const* d_in, const int* in_sizes, int n_in,
                              void* d_out, int out_size, void* d_ws,
                              size_t ws_size, hipStream_t stream) {
  (void)in_sizes; (void)n_in; (void)d_ws; (void)ws_size; (void)out_size;
  const float* img = (const float*)d_in[0];   // (4, 2, 128,128,128) f32
  const float* flow = (const float*)d_in[1];  // (4, 3, 128,128,128) f32
  float* out = (float*)d_out;                 // (4, 2, 128,128,128) f32

  const unsigned totalPoints = NB * NSP;                       // 8,388,608
  const unsigned grid = totalPoints / (TPB * TILES_PER_BLOCK); // 4096 blocks
  st3d_trilinear_kernel<<<dim3(grid), dim3(TPB), 0, stream>>>(img, flow, out);
}